// TripletLoss_36069135351986
// MI455X (gfx1250) — compile-verified
//
#include <hip/hip_runtime.h>

typedef __attribute__((ext_vector_type(2))) float v2f;
typedef __attribute__((ext_vector_type(4))) float v4f;
typedef __attribute__((ext_vector_type(8))) float v8f;

#define TL_MARGIN 0.3f
#define TL_D 128

// ---------------- Kernel 1: row-wise L2 normalization ----------------
// 1 wave (32 lanes) per row; each lane loads float4 (32*16B = 512B = one row).
__global__ __launch_bounds__(128) void tl_normalize(const float* __restrict__ emb,
                                                    float* __restrict__ eN, int n) {
    const int wave = threadIdx.x >> 5;
    const int lane = threadIdx.x & 31;
    const int row  = blockIdx.x * 4 + wave;
    if (row >= n) return;
    v4f x = ((const v4f*)(emb + (size_t)row * TL_D))[lane];
    float ss = x.x * x.x + x.y * x.y + x.z * x.z + x.w * x.w;
    ss += __shfl_xor(ss, 1, 32);
    ss += __shfl_xor(ss, 2, 32);
    ss += __shfl_xor(ss, 4, 32);
    ss += __shfl_xor(ss, 8, 32);
    ss += __shfl_xor(ss, 16, 32);
    float scale = 1.0f / fmaxf(sqrtf(ss), 1e-12f);
    v4f y = {x.x * scale, x.y * scale, x.z * scale, x.w * scale};
    ((v4f*)(eN + (size_t)row * TL_D))[lane] = y;
}

// ---------------- Kernel 2: zero the accumulators ----------------
__global__ void tl_zero(float* acc2) {
    acc2[0] = 0.0f;
    acc2[1] = 0.0f;
}

// ---------------- Kernel 3: fused Gram (WMMA f32) + batch-hard mining ----------
// Block = 128 threads = 4 waves. Wave w owns rows [blockIdx*64 + w*16, +16).
// Per loop iteration: 2 column tiles x 2 K-parity accumulator chains
// => 4 independent v_wmma_f32_16x16x4_f32 dependency chains for latency hiding.
__global__ __launch_bounds__(128) void tl_mine(const float* __restrict__ eN,
                                               const int* __restrict__ labels,
                                               float* __restrict__ acc2, int n) {
    const int wave = threadIdx.x >> 5;
    const int lane = threadIdx.x & 31;
    const int li   = lane & 15;   // matrix row/col index carried by this lane
    const int half = lane >> 4;   // 0: K=v, 1: K=v+2 ; also C-half (M +8)
    const int kh   = half * 2;
    const int rowBase = blockIdx.x * 64 + wave * 16;

    // Preload A panel (16 x 128) in WMMA operand layout: 32 chunks of K=4.
    v2f a[32];
    const float* arow = eN + (size_t)(rowBase + li) * TL_D;
#pragma unroll
    for (int kc = 0; kc < 32; ++kc)
        a[kc] = *(const v2f*)(arow + kc * 4 + kh);

    // Labels of the 16 rows as seen through the C-matrix layout:
    // VGPR r, this lane -> M = r + 8*half.
    int rlab[8];
#pragma unroll
    for (int r = 0; r < 8; ++r)
        rlab[r] = labels[rowBase + r + 8 * half];

    float hpos[8], hneg[8];
#pragma unroll
    for (int r = 0; r < 8; ++r) { hpos[r] = -1.0f; hneg[r] = 3.0f; }

    for (int colBase = 0; colBase < n; colBase += 32) {
        const float* brow0 = eN + (size_t)(colBase + li) * TL_D;
        const float* brow1 = eN + (size_t)(colBase + 16 + li) * TL_D;
        v8f c0a = {0.f, 0.f, 0.f, 0.f, 0.f, 0.f, 0.f, 0.f};
        v8f c0b = {0.f, 0.f, 0.f, 0.f, 0.f, 0.f, 0.f, 0.f};
        v8f c1a = {0.f, 0.f, 0.f, 0.f, 0.f, 0.f, 0.f, 0.f};
        v8f c1b = {0.f, 0.f, 0.f, 0.f, 0.f, 0.f, 0.f, 0.f};
#pragma unroll
        for (int kc = 0; kc < 32; kc += 2) {
            v2f b00 = *(const v2f*)(brow0 + kc * 4 + kh);
            v2f b01 = *(const v2f*)(brow0 + (kc + 1) * 4 + kh);
            v2f b10 = *(const v2f*)(brow1 + kc * 4 + kh);
            v2f b11 = *(const v2f*)(brow1 + (kc + 1) * 4 + kh);
            c0a = __builtin_amdgcn_wmma_f32_16x16x4_f32(
                      false, a[kc],     false, b00, (short)0, c0a, false, false);
            c1a = __builtin_amdgcn_wmma_f32_16x16x4_f32(
                      false, a[kc],     false, b10, (short)0, c1a, false, false);
            c0b = __builtin_amdgcn_wmma_f32_16x16x4_f32(
                      false, a[kc + 1], false, b01, (short)0, c0b, false, false);
            c1b = __builtin_amdgcn_wmma_f32_16x16x4_f32(
                      false, a[kc + 1], false, b11, (short)0, c1b, false, false);
        }
        const int clab0 = labels[colBase + li];
        const int clab1 = labels[colBase + 16 + li];
        const int gcol0 = colBase + li;
        const int gcol1 = colBase + 16 + li;
#pragma unroll
        for (int r = 0; r < 8; ++r) {
            const int grow = rowBase + r + 8 * half;
            // tile 0
            {
                float dist = 1.0f - (c0a[r] + c0b[r]);
                bool same  = (rlab[r] == clab0);
                bool diag  = (grow == gcol0);
                hpos[r] = (same && !diag) ? fmaxf(hpos[r], dist) : hpos[r];
                hneg[r] = (!same)         ? fminf(hneg[r], dist) : hneg[r];
            }
            // tile 1
            {
                float dist = 1.0f - (c1a[r] + c1b[r]);
                bool same  = (rlab[r] == clab1);
                bool diag  = (grow == gcol1);
                hpos[r] = (same && !diag) ? fmaxf(hpos[r], dist) : hpos[r];
                hneg[r] = (!same)         ? fminf(hneg[r], dist) : hneg[r];
            }
        }
    }

    // Reduce over the 16 lane-columns inside each half (xor masks 1..8 stay
    // within a 16-lane group of the wave32).
#pragma unroll
    for (int r = 0; r < 8; ++r) {
#pragma unroll
        for (int m = 1; m <= 8; m <<= 1) {
            hpos[r] = fmaxf(hpos[r], __shfl_xor(hpos[r], m, 32));
            hneg[r] = fminf(hneg[r], __shfl_xor(hneg[r], m, 32));
        }
    }

    if (li == 0) {  // lanes 0 (rows 0..7) and 16 (rows 8..15)
        float lsum = 0.0f, lcnt = 0.0f;
#pragma unroll
        for (int r = 0; r < 8; ++r) {
            // dist in (-eps, 2+eps): sentinel -1 / 3 survive only if no pos/neg.
            bool valid = (hpos[r] > -1.0f) && (hneg[r] < 3.0f);
            float per  = fmaxf(hpos[r] - hneg[r] + TL_MARGIN, 0.0f);
            lsum += valid ? per : 0.0f;
            lcnt += valid ? 1.0f : 0.0f;
        }
        atomicAdd(&acc2[0], lsum);
        atomicAdd(&acc2[1], lcnt);
    }
}

// ---------------- Kernel 4: finalize ----------------
__global__ void tl_finalize(const float* __restrict__ acc2, float* __restrict__ out) {
    out[0] = acc2[0] / fmaxf(acc2[1], 1.0f);
}

extern "C" void kernel_launch(void* const* d_in, const int* in_sizes, int n_in,
                              void* d_out, int out_size, void* d_ws, size_t ws_size,
                              hipStream_t stream) {
    const float* emb  = (const float*)d_in[0];
    const int* labels = (const int*)d_in[1];
    const int n = in_sizes[1];           // 4096 rows (D fixed at 128)

    float* ws   = (float*)d_ws;
    float* acc2 = ws;                    // [0]=loss sum, [1]=valid count
    float* eN   = ws + 16;               // 64B-aligned normalized embeddings

    tl_zero<<<1, 1, 0, stream>>>(acc2);
    tl_normalize<<<(n + 3) / 4, 128, 0, stream>>>(emb, eN, n);
    tl_mine<<<n / 64, 128, 0, stream>>>(eN, labels, acc2, n);
    tl_finalize<<<1, 1, 0, stream>>>(acc2, (float*)d_out);
}